// ProteinFeatures_87514253623869
// MI455X (gfx1250) — compile-verified
//
#include <hip/hip_runtime.h>
#include <hip/hip_bf16.h>
#include <math.h>

typedef __attribute__((ext_vector_type(2))) float v2f;
typedef __attribute__((ext_vector_type(8))) float v8f;

#define Bb 8
#define Nn 2048
#define Kn 30
#define NFe 128
#define MAXRELc 32
#define EW 2          // waves per workgroup in MLP kernels

__device__ __forceinline__ float3 f3(float x, float y, float z) { return make_float3(x, y, z); }
__device__ __forceinline__ float3 sub3(float3 a, float3 b) { return f3(a.x - b.x, a.y - b.y, a.z - b.z); }
__device__ __forceinline__ float dot3(float3 a, float3 b) { return a.x * b.x + a.y * b.y + a.z * b.z; }
__device__ __forceinline__ float3 cross3(float3 a, float3 b) {
    return f3(a.y * b.z - a.z * b.y, a.z * b.x - a.x * b.z, a.x * b.y - a.y * b.x);
}
__device__ __forceinline__ float3 norm3(float3 a) {
    float n = sqrtf(dot3(a, a));
    float s = 1.f / (n + 1e-8f);
    return f3(a.x * s, a.y * s, a.z * s);
}
__device__ __forceinline__ float sgn(float x) { return x > 0.f ? 1.f : (x < 0.f ? -1.f : 0.f); }

__device__ __forceinline__ v8f vzero8() {
    v8f z = {0.f, 0.f, 0.f, 0.f, 0.f, 0.f, 0.f, 0.f};
    return z;
}
__device__ __forceinline__ v8f wmma_f32(v2f a, v2f b, v8f c) {
    // D(16x16 f32) = A(16x4 f32) * B(4x16 f32) + C
    return __builtin_amdgcn_wmma_f32_16x16x4_f32(false, a, false, b, (short)0, c, false, false);
}

// ---------------------------------------------------------------------------
// Kernel 1: per-node dihedral features (6, padded to 8) and orientation frame O (9)
// ---------------------------------------------------------------------------
__global__ void geom_kernel(const float* __restrict__ X, float* __restrict__ O,
                            float* __restrict__ Vf) {
    int row = blockIdx.x * 256 + threadIdx.x;
    if (row >= Bb * Nn) return;
    int b = row / Nn, n = row % Nn;
    const float* Xb = X + (size_t)b * Nn * 12;

    float cs[3], sn[3];
    for (int a = 0; a < 3; a++) {
        int t = 3 * n + a;
        float ang = 0.f;
        if (t >= 1 && t < 3 * Nn - 2) {
            int m = t - 1;
            float3 P[4];
            for (int q = 0; q < 4; q++) {
                int g = m + q;
                const float* p = Xb + (g / 3) * 12 + (g % 3) * 3;
                P[q] = f3(p[0], p[1], p[2]);
            }
            float3 u2 = norm3(sub3(P[1], P[0]));
            float3 u1 = norm3(sub3(P[2], P[1]));
            float3 u0 = norm3(sub3(P[3], P[2]));
            float3 n2 = norm3(cross3(u2, u1));
            float3 n1 = norm3(cross3(u1, u0));
            float cD = fminf(fmaxf(dot3(n2, n1), -1.f + 1e-7f), 1.f - 1e-7f);
            ang = sgn(dot3(u2, n1)) * acosf(cD);
        }
        cs[a] = cosf(ang);
        sn[a] = sinf(ang);
    }
    float* vf = Vf + (size_t)row * 8;
    vf[0] = cs[0]; vf[1] = cs[1]; vf[2] = cs[2];
    vf[3] = sn[0]; vf[4] = sn[1]; vf[5] = sn[2];
    vf[6] = 0.f; vf[7] = 0.f;

    float* o = O + (size_t)row * 9;
    if (n >= 1 && n <= Nn - 3) {
        int i = n - 1;
        const float* c0 = Xb + i * 12 + 3;
        const float* c1 = Xb + (i + 1) * 12 + 3;
        const float* c2 = Xb + (i + 2) * 12 + 3;
        float3 A0 = f3(c0[0], c0[1], c0[2]);
        float3 A1 = f3(c1[0], c1[1], c1[2]);
        float3 A2 = f3(c2[0], c2[1], c2[2]);
        float3 u2 = norm3(sub3(A1, A0));
        float3 u1 = norm3(sub3(A2, A1));
        float3 n2 = norm3(cross3(u2, u1));
        float3 o1 = norm3(sub3(u2, u1));
        float3 o3 = cross3(o1, n2);
        o[0] = o1.x; o[1] = o1.y; o[2] = o1.z;
        o[3] = n2.x; o[4] = n2.y; o[5] = n2.z;
        o[6] = o3.x; o[7] = o3.y; o[8] = o3.z;
    } else {
        for (int q = 0; q < 9; q++) o[q] = 0.f;
    }
}

// ---------------------------------------------------------------------------
// Kernel 2: pairwise Ca distances + stable top-K(=30) smallest per row
// ---------------------------------------------------------------------------
#define TKT 64
#define DTILE 256
__global__ void topk_kernel(const float* __restrict__ X, const float* __restrict__ mask,
                            float* __restrict__ Dnb, int* __restrict__ Eidx) {
    __shared__ float sc[DTILE * 3];
    __shared__ float sm[DTILE];
    __shared__ float sval[TKT][Kn];
    __shared__ int sidx[TKT][Kn];
    int tid = threadIdx.x;
    int row = blockIdx.x * TKT + tid;
    int b = row / Nn, i = row % Nn;
    const float* Xb = X + (size_t)b * Nn * 12;
    const float* ci = Xb + i * 12 + 3;
    float cix = ci[0], ciy = ci[1], ciz = ci[2];
    float mi = mask[row];
    int cnt = 0;
    for (int j0 = 0; j0 < Nn; j0 += DTILE) {
        __syncthreads();
        for (int q = tid; q < DTILE; q += TKT) {
            const float* cj = Xb + (size_t)(j0 + q) * 12 + 3;
            sc[q * 3 + 0] = cj[0];
            sc[q * 3 + 1] = cj[1];
            sc[q * 3 + 2] = cj[2];
            sm[q] = mask[b * Nn + j0 + q];
        }
        __syncthreads();
        for (int q = 0; q < DTILE; q++) {
            float dx = cix - sc[q * 3], dy = ciy - sc[q * 3 + 1], dz = ciz - sc[q * 3 + 2];
            float d = sqrtf(dx * dx + dy * dy + dz * dz + 1e-6f) + (1.f - mi * sm[q]) * 10000.f;
            int j = j0 + q;
            if (cnt < Kn || d < sval[tid][Kn - 1]) {
                int p = (cnt < Kn) ? cnt : (Kn - 1);
                while (p > 0 && sval[tid][p - 1] > d) {
                    sval[tid][p] = sval[tid][p - 1];
                    sidx[tid][p] = sidx[tid][p - 1];
                    p--;
                }
                sval[tid][p] = d;
                sidx[tid][p] = j;
                if (cnt < Kn) cnt++;
            }
        }
    }
    for (int k = 0; k < Kn; k++) {
        Dnb[(size_t)row * Kn + k] = sval[tid][k];
        Eidx[(size_t)row * Kn + k] = sidx[tid][k];
    }
}

// ---------------------------------------------------------------------------
// Kernel 2.5: pair-swizzle K dim of 128x128 weights so each WMMA B fragment is
// one contiguous 8-byte load: W2s[(k>>1)*256 + c*2 + (k&1)] = W2[k*128 + c]
// ---------------------------------------------------------------------------
__global__ void swizzle_w2_kernel(const float* __restrict__ W2e, const float* __restrict__ W2n,
                                  float* __restrict__ W2es, float* __restrict__ W2ns) {
    int q = blockIdx.x * 256 + threadIdx.x;   // 0 .. 16383
    int k = q >> 7, c = q & 127;
    int d = (k >> 1) * 256 + c * 2 + (k & 1);
    W2es[d] = W2e[q];
    W2ns[d] = W2n[q];
}

// ---------------------------------------------------------------------------
// Kernel 3: edge features (39, pad 40) + [40->128] + [128->128] + LayerNorm, WMMA f32
// ---------------------------------------------------------------------------
__global__ void edge_mlp_kernel(const float* __restrict__ X, const int* __restrict__ ridx,
                                const int* __restrict__ chain, const float* __restrict__ pe_w,
                                const float* __restrict__ pe_b, const float* __restrict__ W1,
                                const float* __restrict__ b1, const float* __restrict__ W2s,
                                const float* __restrict__ b2, const float* __restrict__ g,
                                const float* __restrict__ beta, const float* __restrict__ O,
                                const float* __restrict__ Dnb, const int* __restrict__ Eidx,
                                float* __restrict__ Eout) {
    __shared__ float sW1[20 * 256];           // pair-swizzled 40x128
    __shared__ float sB1[128], sB2[128], sG[128], sBt[128];
    __shared__ float sEA[EW][16 * 40];
    __shared__ float sH[EW][16 * 128];
    __shared__ float sH2[EW][16 * 128];
    int tid = threadIdx.x;
    int wave = tid >> 5, lane = tid & 31;

    __builtin_prefetch(W2s + ((tid * 64) & 16383), 0, 1);

    // stage W1 pair-swizzled: sW1[(k>>1)*256 + c*2 + (k&1)], rows >= 39 zero
    for (int q = tid; q < 40 * 128; q += EW * 32) {
        int k = q >> 7, c = q & 127;
        sW1[(k >> 1) * 256 + c * 2 + (k & 1)] = (k < 39) ? W1[q] : 0.f;
    }
    for (int q = tid; q < 128; q += EW * 32) {
        sB1[q] = b1[q]; sB2[q] = b2[q]; sG[q] = g[q]; sBt[q] = beta[q];
    }
    __syncthreads();

    int tile = blockIdx.x * EW + wave;
    int e0 = tile * 16;

    if (lane < 16) {
        int e = e0 + lane;
        int b = e / (Nn * Kn);
        int rem = e % (Nn * Kn);
        int i = rem / Kn;
        int j = Eidx[e];
        float* EA = &sEA[wave][lane * 40];
        // positional encoding: one-hot @ pe_w == row gather
        int off = ridx[b * Nn + i] - ridx[b * Nn + j];
        int ch = (chain[b * Nn + i] == chain[b * Nn + j]) ? 1 : 0;
        int dd = off + MAXRELc;
        dd = dd < 0 ? 0 : (dd > 2 * MAXRELc ? 2 * MAXRELc : dd);
        int sel = ch ? dd : (2 * MAXRELc + 1);
        for (int c = 0; c < 16; c++) EA[c] = pe_w[sel * 16 + c] + pe_b[c];
        // RBF
        float dn = Dnb[e];
        for (int c = 0; c < 16; c++) {
            float mu = 2.f + (20.f / 15.f) * (float)c;
            float z = (dn - mu) * (1.f / 1.25f);
            EA[16 + c] = expf(-z * z);
        }
        // orientation features
        const float* Of = O + (size_t)(b * Nn + i) * 9;
        const float* On = O + (size_t)(b * Nn + j) * 9;
        const float* cI = X + ((size_t)(b * Nn + i) * 4 + 1) * 3;
        const float* cJ = X + ((size_t)(b * Nn + j) * 4 + 1) * 3;
        float3 dX = f3(cJ[0] - cI[0], cJ[1] - cI[1], cJ[2] - cI[2]);
        float3 du = f3(Of[0] * dX.x + Of[1] * dX.y + Of[2] * dX.z,
                       Of[3] * dX.x + Of[4] * dX.y + Of[5] * dX.z,
                       Of[6] * dX.x + Of[7] * dX.y + Of[8] * dX.z);
        du = norm3(du);
        float R[3][3];
        for (int r = 0; r < 3; r++)
            for (int c = 0; c < 3; c++)
                R[r][c] = Of[0 + r] * On[0 + c] + Of[3 + r] * On[3 + c] + Of[6 + r] * On[6 + c];
        float Rxx = R[0][0], Ryy = R[1][1], Rzz = R[2][2];
        float m0 = 0.5f * sqrtf(fabsf(1.f + Rxx - Ryy - Rzz));
        float m1 = 0.5f * sqrtf(fabsf(1.f - Rxx + Ryy - Rzz));
        float m2 = 0.5f * sqrtf(fabsf(1.f - Rxx - Ryy + Rzz));
        float q0 = sgn(R[2][1] - R[1][2]) * m0;
        float q1 = sgn(R[0][2] - R[2][0]) * m1;
        float q2 = sgn(R[1][0] - R[0][1]) * m2;
        float w = 0.5f * sqrtf(fmaxf(0.f, 1.f + Rxx + Ryy + Rzz));
        float qs = 1.f / (sqrtf(q0 * q0 + q1 * q1 + q2 * q2 + w * w) + 1e-8f);
        EA[32] = du.x; EA[33] = du.y; EA[34] = du.z;
        EA[35] = q0 * qs; EA[36] = q1 * qs; EA[37] = q2 * qs; EA[38] = w * qs;
        EA[39] = 0.f;
    }
    __syncthreads();

    int half = lane >> 4;
    int l15 = lane & 15;

    // GEMM1: (16x40) x (40x128); all fragments are single aligned v2f loads
    v8f acc[8];
#pragma unroll
    for (int t = 0; t < 8; t++) acc[t] = vzero8();
    for (int kk = 0; kk < 10; kk++) {
        v2f a = *(const v2f*)&sEA[wave][l15 * 40 + kk * 4 + half * 2];
        int p = 2 * kk + half;
#pragma unroll
        for (int t = 0; t < 8; t++) {
            v2f bb = *(const v2f*)&sW1[p * 256 + (t * 16 + l15) * 2];
            acc[t] = wmma_f32(a, bb, acc[t]);
        }
    }
#pragma unroll
    for (int t = 0; t < 8; t++) {
#pragma unroll
        for (int v = 0; v < 8; v++) {
            sH[wave][(v + 8 * half) * 128 + t * 16 + l15] = acc[t][v] + sB1[t * 16 + l15];
        }
    }
    __syncthreads();

    // GEMM2: (16x128) x (128x128); W2 pair-swizzled in global (L2-resident, 64 KB)
#pragma unroll
    for (int t = 0; t < 8; t++) acc[t] = vzero8();
    for (int kk = 0; kk < 32; kk++) {
        v2f a = *(const v2f*)&sH[wave][l15 * 128 + kk * 4 + half * 2];
        int p = 2 * kk + half;
#pragma unroll
        for (int t = 0; t < 8; t++) {
            v2f bb = *(const v2f*)&W2s[p * 256 + (t * 16 + l15) * 2];
            acc[t] = wmma_f32(a, bb, acc[t]);
        }
    }
#pragma unroll
    for (int t = 0; t < 8; t++) {
#pragma unroll
        for (int v = 0; v < 8; v++) {
            sH2[wave][(v + 8 * half) * 128 + t * 16 + l15] = acc[t][v] + sB2[t * 16 + l15];
        }
    }
    __syncthreads();

    if (lane < 16) {
        const float4* hr4 = (const float4*)&sH2[wave][lane * 128];
        float mu = 0.f;
        for (int c = 0; c < 32; c++) {
            float4 h = hr4[c];
            mu += h.x + h.y + h.z + h.w;
        }
        mu *= (1.f / 128.f);
        float var = 0.f;
        for (int c = 0; c < 32; c++) {
            float4 h = hr4[c];
            float dx = h.x - mu, dy = h.y - mu, dz = h.z - mu, dw = h.w - mu;
            var += dx * dx + dy * dy + dz * dz + dw * dw;
        }
        var *= (1.f / 128.f);
        float rs = rsqrtf(var + 1e-5f);
        float4* outp = (float4*)(Eout + (size_t)(e0 + lane) * 128);
        const float4* g4 = (const float4*)sG;
        const float4* bt4 = (const float4*)sBt;
        for (int c = 0; c < 32; c++) {
            float4 h = hr4[c], gg = g4[c], bb = bt4[c], o;
            o.x = (h.x - mu) * rs * gg.x + bb.x;
            o.y = (h.y - mu) * rs * gg.y + bb.y;
            o.z = (h.z - mu) * rs * gg.z + bb.z;
            o.w = (h.w - mu) * rs * gg.w + bb.w;
            outp[c] = o;
        }
    }
}

// ---------------------------------------------------------------------------
// Kernel 4: node MLP [8->128] + [128->128] + LayerNorm, WMMA f32
// ---------------------------------------------------------------------------
__global__ void node_mlp_kernel(const float* __restrict__ Vf, const float* __restrict__ W1,
                                const float* __restrict__ b1, const float* __restrict__ W2s,
                                const float* __restrict__ b2, const float* __restrict__ g,
                                const float* __restrict__ beta, float* __restrict__ Vout) {
    __shared__ float sW1[4 * 256];            // pair-swizzled 8x128
    __shared__ float sB1[128], sB2[128], sG[128], sBt[128];
    __shared__ float sH[EW][16 * 128];
    __shared__ float sH2[EW][16 * 128];
    int tid = threadIdx.x;
    int wave = tid >> 5, lane = tid & 31;

    __builtin_prefetch(W2s + ((tid * 64) & 16383), 0, 1);

    for (int q = tid; q < 8 * 128; q += EW * 32) {
        int k = q >> 7, c = q & 127;
        sW1[(k >> 1) * 256 + c * 2 + (k & 1)] = (k < 6) ? W1[q] : 0.f;
    }
    for (int q = tid; q < 128; q += EW * 32) {
        sB1[q] = b1[q]; sB2[q] = b2[q]; sG[q] = g[q]; sBt[q] = beta[q];
    }
    __syncthreads();

    int tile = blockIdx.x * EW + wave;
    int r0 = tile * 16;
    int half = lane >> 4;
    int l15 = lane & 15;

    v8f acc[8];
#pragma unroll
    for (int t = 0; t < 8; t++) acc[t] = vzero8();
    for (int kk = 0; kk < 2; kk++) {
        v2f a = *(const v2f*)(Vf + (size_t)(r0 + l15) * 8 + kk * 4 + half * 2);
        int p = 2 * kk + half;
#pragma unroll
        for (int t = 0; t < 8; t++) {
            v2f bb = *(const v2f*)&sW1[p * 256 + (t * 16 + l15) * 2];
            acc[t] = wmma_f32(a, bb, acc[t]);
        }
    }
#pragma unroll
    for (int t = 0; t < 8; t++) {
#pragma unroll
        for (int v = 0; v < 8; v++) {
            sH[wave][(v + 8 * half) * 128 + t * 16 + l15] = acc[t][v] + sB1[t * 16 + l15];
        }
    }
    __syncthreads();

#pragma unroll
    for (int t = 0; t < 8; t++) acc[t] = vzero8();
    for (int kk = 0; kk < 32; kk++) {
        v2f a = *(const v2f*)&sH[wave][l15 * 128 + kk * 4 + half * 2];
        int p = 2 * kk + half;
#pragma unroll
        for (int t = 0; t < 8; t++) {
            v2f bb = *(const v2f*)&W2s[p * 256 + (t * 16 + l15) * 2];
            acc[t] = wmma_f32(a, bb, acc[t]);
        }
    }
#pragma unroll
    for (int t = 0; t < 8; t++) {
#pragma unroll
        for (int v = 0; v < 8; v++) {
            sH2[wave][(v + 8 * half) * 128 + t * 16 + l15] = acc[t][v] + sB2[t * 16 + l15];
        }
    }
    __syncthreads();

    if (lane < 16) {
        const float4* hr4 = (const float4*)&sH2[wave][lane * 128];
        float mu = 0.f;
        for (int c = 0; c < 32; c++) {
            float4 h = hr4[c];
            mu += h.x + h.y + h.z + h.w;
        }
        mu *= (1.f / 128.f);
        float var = 0.f;
        for (int c = 0; c < 32; c++) {
            float4 h = hr4[c];
            float dx = h.x - mu, dy = h.y - mu, dz = h.z - mu, dw = h.w - mu;
            var += dx * dx + dy * dy + dz * dz + dw * dw;
        }
        var *= (1.f / 128.f);
        float rs = rsqrtf(var + 1e-5f);
        float4* outp = (float4*)(Vout + (size_t)(r0 + lane) * 128);
        const float4* g4 = (const float4*)sG;
        const float4* bt4 = (const float4*)sBt;
        for (int c = 0; c < 32; c++) {
            float4 h = hr4[c], gg = g4[c], bb = bt4[c], o;
            o.x = (h.x - mu) * rs * gg.x + bb.x;
            o.y = (h.y - mu) * rs * gg.y + bb.y;
            o.z = (h.z - mu) * rs * gg.z + bb.z;
            o.w = (h.w - mu) * rs * gg.w + bb.w;
            outp[c] = o;
        }
    }
}

// ---------------------------------------------------------------------------
extern "C" void kernel_launch(void* const* d_in, const int* in_sizes, int n_in, void* d_out,
                              int out_size, void* d_ws, size_t ws_size, hipStream_t stream) {
    (void)in_sizes; (void)n_in; (void)out_size; (void)ws_size;
    const float* X = (const float*)d_in[0];
    const float* mask = (const float*)d_in[1];
    const int* ridx = (const int*)d_in[2];
    const int* chain = (const int*)d_in[3];
    const float* pe_w = (const float*)d_in[4];
    const float* pe_b = (const float*)d_in[5];
    const float* nlw = (const float*)d_in[6];
    const float* nlb = (const float*)d_in[7];
    const float* nemw = (const float*)d_in[8];
    const float* nemb = (const float*)d_in[9];
    const float* elw = (const float*)d_in[10];
    const float* elb = (const float*)d_in[11];
    const float* eew = (const float*)d_in[12];
    const float* eeb = (const float*)d_in[13];
    const float* nng = (const float*)d_in[14];
    const float* nnb = (const float*)d_in[15];
    const float* neg = (const float*)d_in[16];
    const float* nebt = (const float*)d_in[17];

    float* out = (float*)d_out;
    float* Vout = out;                                        // B*N*128
    float* Eout = out + (size_t)Bb * Nn * NFe;                // B*N*K*128
    int* EidxO = (int*)(Eout + (size_t)Bb * Nn * Kn * NFe);   // B*N*K (int32)

    float* ws = (float*)d_ws;
    float* O = ws;                            // B*N*9
    float* Vf = O + (size_t)Bb * Nn * 9;      // B*N*8 (padded node feats)
    float* Dnb = Vf + (size_t)Bb * Nn * 8;    // B*N*K
    float* W2es = Dnb + (size_t)Bb * Nn * Kn; // 128*128 swizzled edge_emb_w
    float* W2ns = W2es + 128 * 128;           // 128*128 swizzled node_emb_w

    geom_kernel<<<(Bb * Nn) / 256, 256, 0, stream>>>(X, O, Vf);
    topk_kernel<<<(Bb * Nn) / TKT, TKT, 0, stream>>>(X, mask, Dnb, EidxO);
    swizzle_w2_kernel<<<64, 256, 0, stream>>>(eew, nemw, W2es, W2ns);
    node_mlp_kernel<<<((Bb * Nn) / 16) / EW, EW * 32, 0, stream>>>(Vf, nlw, nlb, W2ns, nemb,
                                                                   nng, nnb, Vout);
    edge_mlp_kernel<<<((Bb * Nn * Kn) / 16) / EW, EW * 32, 0, stream>>>(
        X, ridx, chain, pe_w, pe_b, elw, elb, W2es, eeb, neg, nebt, O, Dnb, EidxO, Eout);
}